// TimeChannelEstimation_39960375722141
// MI455X (gfx1250) — compile-verified
//
#include <hip/hip_runtime.h>

// TimeChannelEstimation — MI455X (gfx1250, wave32)
//
// Math collapse: fft(ifft(fft(x)*conj(fft(ifft_ortho(zc))))) == fft(x)*conj(zc),
// so  out[b,s,:] = cov[k(b,s)] @ (FFT(x[b,s]) * conj(zc))   (complex matvec).
// Memory-bound: 112 gathered 256x256 complex64 cov matrices ~ 58.7 MB -> ~2.5 us
// floor at 23.3 TB/s. Matvec compute uses V_WMMA_F32_16X16X4_F32 with a
// branchless B-fragment (yr in col 0, yi in col 1, zero plane for cols 2..15).

constexpr int kN   = 256;
constexpr int kB   = 8;
constexpr int kS   = 14;
constexpr int kSym = kB * kS;   // 112

typedef __attribute__((ext_vector_type(2))) float v2f;
typedef __attribute__((ext_vector_type(8))) float v8f;

#if defined(__has_builtin)
#if __has_builtin(__builtin_amdgcn_wmma_f32_16x16x4_f32)
#define HAVE_WMMA_F32X4 1
#endif
#endif

// ---------------------------------------------------------------------------
// Kernel 1: per (b,s) 256-pt DFT (direct, LDS twiddle table) + conj(zc) mult.
// Y stored SoA in workspace: Yr[112*256], Yi[112*256].
// ---------------------------------------------------------------------------
__global__ __launch_bounds__(256) void dft_zc_kernel(
    const float* __restrict__ xr, const float* __restrict__ xi,
    const float* __restrict__ zcr, const float* __restrict__ zci,
    float* __restrict__ Yr, float* __restrict__ Yi)
{
    __shared__ float lxr[kN], lxi[kN], twr[kN], twi[kN];
    const int bs = blockIdx.x;
    const int t  = threadIdx.x;

    lxr[t] = xr[bs * kN + t];
    lxi[t] = xi[bs * kN + t];
    float si, co;
    __sincosf(-6.28318530717958647692f * (float)t / (float)kN, &si, &co);
    twr[t] = co;
    twi[t] = si;
    __syncthreads();

    float ar = 0.f, ai = 0.f;
#pragma unroll 4
    for (int m = 0; m < kN; ++m) {
        const int idx = (t * m) & (kN - 1);
        const float wr = twr[idx], wi = twi[idx];
        const float vr = lxr[m],  vi = lxi[m];
        ar = fmaf(vr, wr, fmaf(-vi, wi, ar));
        ai = fmaf(vr, wi, fmaf( vi, wr, ai));
    }
    // X * conj(zc):  (ar + i*ai)(zr - i*zi)
    const float zr = zcr[t], zi = zci[t];
    Yr[bs * kN + t] = fmaf(ar, zr,  ai * zi);
    Yi[bs * kN + t] = fmaf(ai, zr, -ar * zi);
}

// ---------------------------------------------------------------------------
// cov index: complete = pad(shift, 1 zero col)[:, gather_idx]
//   s in {0,7} -> 0 ; 1<=s<=6 -> shift[b, s-1] ; 8<=s<=13 -> shift[b, s-2]
// ---------------------------------------------------------------------------
__device__ __forceinline__ int cov_index(const int* __restrict__ shift, int b, int s)
{
    if (s == 0 || s == 7) return 0;
    return (s < 7) ? shift[b * 12 + (s - 1)] : shift[b * 12 + (s - 2)];
}

#ifdef HAVE_WMMA_F32X4
// ---------------------------------------------------------------------------
// Kernel 2 (WMMA): grid (112, 4), 256 threads = 8 waves.
//   wave w: n-tile = (w&3), K-half = (w>>2)  -> n0 = (4*by + (w&3))*16,
//           m in [128*(w>>2), 128*(w>>2)+128)  (32 K-steps of 4).
// A-frag (16x4 f32, 2 VGPR): lane&15 = M row, lane>>4 picks K-pair.
// B-frag (4x16 f32, 2 VGPR): lane&15 = N col, lane>>4 picks K-pair.
//   LDS ybuf = [ yr(256) | yi(256) | zeros(256) ]; every lane does the same
//   ds_load_b64 from plane p = min(col,2) -> branchless, EXEC stays all-ones.
// accR (A=cov_re): col0 = Cr.yr, col1 = Cr.yi
// accI (A=cov_im): col0 = Ci.yr, col1 = Ci.yi
// out_re = accR.c0 - accI.c1 ; out_im = accR.c1 + accI.c0  (K-halves summed).
// ---------------------------------------------------------------------------
__global__ __launch_bounds__(256) void matvec_wmma_kernel(
    const float* __restrict__ covR, const float* __restrict__ covI,
    const int* __restrict__ shift,
    const float* __restrict__ Yr, const float* __restrict__ Yi,
    float* __restrict__ out)
{
    __shared__ float ybuf[3 * kN];          // yr | yi | zero plane
    __shared__ float scR[8][2][16];
    __shared__ float scI[8][2][16];

    const int bs = blockIdx.x;
    const int b  = bs / kS;
    const int s  = bs % kS;
    const int k  = cov_index(shift, b, s);

    const int t = threadIdx.x;
    ybuf[t]            = Yr[bs * kN + t];
    ybuf[kN + t]       = Yi[bs * kN + t];
    ybuf[2 * kN + t]   = 0.f;
    __syncthreads();

    const int lane  = t & 31;
    const int w     = t >> 5;
    const int nc    = lane & 15;            // M for A-frag, N for B-frag
    const int kh    = lane >> 4;            // K-pair select within a step
    const int ntile = w & 3;
    const int kbase = (w >> 2) * 128;       // K-half of the m dimension
    const int n0    = (blockIdx.y * 4 + ntile) * 16;

    const size_t covOff = ((size_t)k << 16) + (size_t)(n0 + nc) * kN + kbase + 2 * kh;
    const float* Ar = covR + covOff;
    const float* Ai = covI + covOff;

    const int p = (nc < 2) ? nc : 2;        // B plane: yr / yi / zeros
    const float* pB = &ybuf[p * kN + kbase + 2 * kh];

    v8f accR0 = {}, accR1 = {}, accI0 = {}, accI1 = {};

#pragma unroll 4
    for (int ks = 0; ks < 32; ks += 2) {
        {
            const int mb = ks * 4;
            const v2f aR = *(const v2f*)(Ar + mb);
            const v2f aI = *(const v2f*)(Ai + mb);
            const v2f bf = *(const v2f*)(pB + mb);
            accR0 = __builtin_amdgcn_wmma_f32_16x16x4_f32(
                false, aR, false, bf, (short)0, accR0, false, false);
            accI0 = __builtin_amdgcn_wmma_f32_16x16x4_f32(
                false, aI, false, bf, (short)0, accI0, false, false);
        }
        {
            const int mb = (ks + 1) * 4;
            const v2f aR = *(const v2f*)(Ar + mb);
            const v2f aI = *(const v2f*)(Ai + mb);
            const v2f bf = *(const v2f*)(pB + mb);
            accR1 = __builtin_amdgcn_wmma_f32_16x16x4_f32(
                false, aR, false, bf, (short)0, accR1, false, false);
            accI1 = __builtin_amdgcn_wmma_f32_16x16x4_f32(
                false, aI, false, bf, (short)0, accI1, false, false);
        }
    }

    const v8f accR = accR0 + accR1;
    const v8f accI = accI0 + accI1;

    // D layout: VGPR j, lanes 0-15 -> (M=j, N=lane); lanes 16-31 -> (M=8+j).
    // Only columns N=0 (lanes 0,16) and N=1 (lanes 1,17) are meaningful.
    if (nc < 2) {
#pragma unroll
        for (int j = 0; j < 8; ++j) {
            scR[w][nc][kh * 8 + j] = accR[j];
            scI[w][nc][kh * 8 + j] = accI[j];
        }
    }
    __syncthreads();

    // Waves 0-3 combine their K-half with waves 4-7 and emit complex output.
    if (w < 4 && lane < 16) {
        const float r0 = scR[w][0][lane] + scR[w + 4][0][lane];
        const float r1 = scR[w][1][lane] + scR[w + 4][1][lane];
        const float i0 = scI[w][0][lane] + scI[w + 4][0][lane];
        const float i1 = scI[w][1][lane] + scI[w + 4][1][lane];
        const int n = n0 + lane;
        float2 v = make_float2(r0 - i1, r1 + i0);
        *(float2*)(out + (((size_t)bs * kN + n) << 1)) = v;  // interleaved complex64
    }
}
#else
// ---------------------------------------------------------------------------
// Fallback (no f32 WMMA builtin): plain VALU matvec, one row per thread.
// ---------------------------------------------------------------------------
__global__ __launch_bounds__(256) void matvec_plain_kernel(
    const float* __restrict__ covR, const float* __restrict__ covI,
    const int* __restrict__ shift,
    const float* __restrict__ Yr, const float* __restrict__ Yi,
    float* __restrict__ out)
{
    __shared__ float yr[kN], yi[kN];
    const int bs = blockIdx.x;
    const int b  = bs / kS;
    const int s  = bs % kS;
    const int k  = cov_index(shift, b, s);
    const int n  = threadIdx.x;

    yr[n] = Yr[bs * kN + n];
    yi[n] = Yi[bs * kN + n];
    __syncthreads();

    const float* cr = covR + ((size_t)k << 16) + (size_t)n * kN;
    const float* ci = covI + ((size_t)k << 16) + (size_t)n * kN;
    float re = 0.f, im = 0.f;
#pragma unroll 4
    for (int m = 0; m < kN; ++m) {
        const float r = cr[m], i = ci[m];
        re = fmaf(r, yr[m], fmaf(-i, yi[m], re));
        im = fmaf(r, yi[m], fmaf( i, yr[m], im));
    }
    float2 v = make_float2(re, im);
    *(float2*)(out + (((size_t)bs * kN + n) << 1)) = v;
}
#endif

// ---------------------------------------------------------------------------
extern "C" void kernel_launch(void* const* d_in, const int* in_sizes, int n_in,
                              void* d_out, int out_size, void* d_ws, size_t ws_size,
                              hipStream_t stream)
{
    const float* xr    = (const float*)d_in[0];   // (B,S,N)
    const float* xi    = (const float*)d_in[1];
    const float* covR  = (const float*)d_in[2];   // (N,N,N)
    const float* covI  = (const float*)d_in[3];
    const float* zcr   = (const float*)d_in[4];   // (N)
    const float* zci   = (const float*)d_in[5];
    const int*   shift = (const int*)d_in[6];     // (B, S-2)
    (void)in_sizes; (void)n_in; (void)out_size; (void)ws_size;
    // d_in[7] (gather_idx) is a compile-time-fixed permutation; baked into cov_index().

    float* Yr  = (float*)d_ws;
    float* Yi  = Yr + (size_t)kSym * kN;
    float* out = (float*)d_out;                   // complex64 interleaved

    dft_zc_kernel<<<dim3(kSym), dim3(256), 0, stream>>>(xr, xi, zcr, zci, Yr, Yi);
#ifdef HAVE_WMMA_F32X4
    matvec_wmma_kernel<<<dim3(kSym, 4), dim3(256), 0, stream>>>(covR, covI, shift, Yr, Yi, out);
#else
    matvec_plain_kernel<<<dim3(kSym), dim3(256), 0, stream>>>(covR, covI, shift, Yr, Yi, out);
#endif
}